// GetNodeK_61332132987194
// MI455X (gfx1250) — compile-verified
//
#include <hip/hip_runtime.h>
#include <cstdint>

#define NFEAT 128
#define BATCH 4
#define AT    512
#define NBR   32
#define NBR1  31   /* NBR - 1 */

typedef float v4f __attribute__((ext_vector_type(4)));

// One block per (b, a). 256 threads = 8 wave32.
// Phase 1: async-stage the 32 unique gathered rows (16 KB) into LDS.
// Phase 2: each wave streams one 512-B output row per iteration:
//          lane l moves one float4 -> fully coalesced b128 NT stores.
__global__ __launch_bounds__(256) void GetNodeK_gather_kernel(
    const float* __restrict__ emb,   // (B, AT, 128) f32
    const int*   __restrict__ nbr,   // (B, AT, 32)  int
    float*       __restrict__ out)   // (B, AT, 32, 31, 128) f32
{
    __shared__ float srows[NBR * NFEAT];  // 32 rows x 512 B = 16 KB
    __shared__ int   sidx[NBR];

    const int ba   = blockIdx.x;          // 0..2047 => (b, a)
    const int b    = ba >> 9;             // ba / AT
    const int tid  = threadIdx.x;         // 0..255

    // Load the 32 neighbor indices for this (b, a).
    if (tid < NBR) {
        sidx[tid] = nbr[(size_t)ba * NBR + tid];
    }
    __syncthreads();

    // ---- Phase 1: stage 32 unique embedding rows into LDS ----------------
    // 32 rows * 32 float4 = 1024 b128 transfers; 256 threads * 4 iterations.
    {
        const float* embB = emb + (size_t)b * AT * NFEAT;
#pragma unroll
        for (int t = 0; t < 4; ++t) {
            const int e   = tid + t * 256;   // 0..1023: float4 slot
            const int row = e >> 5;          // 0..31
            const int seg = e & 31;          // float4 within row
            const int r   = sidx[row];       // source row in embedding table

            const uint64_t gaddr =
                (uint64_t)(uintptr_t)(embB + (size_t)r * NFEAT + seg * 4);
            // Flat LDS addresses carry the byte offset in the low 32 bits.
            const uint32_t lds_addr =
                (uint32_t)(uintptr_t)(&srows[row * NFEAT + seg * 4]);

            // CDNA5 async copy global -> LDS (tracked by ASYNCcnt).
            asm volatile("global_load_async_to_lds_b128 %0, %1, off"
                         :
                         : "v"(lds_addr), "v"(gaddr)
                         : "memory");
        }
        // Wait for this wave's async transfers, then cross-wave barrier.
        asm volatile("s_wait_asynccnt 0" ::: "memory");
    }
    __syncthreads();

    // ---- Phase 2: stream 992 rows out of LDS -----------------------------
    float* outBA = out + (size_t)ba * (NBR * NBR1 * NFEAT);
    const int wave = tid >> 5;   // 0..7
    const int lane = tid & 31;   // 0..31  (one float4 per lane)

    // 992 rows / 8 waves = 124 iterations per wave; unroll x4 to keep
    // several b128 loads/stores in flight.
#pragma unroll 4
    for (int p = wave; p < NBR * NBR1; p += 8) {
        const int i = p / NBR1;              // output neighbor slot
        const int m = p - i * NBR1;          // 0..30
        const int k = m + (m >= i ? 1 : 0);  // skip the diagonal

        const v4f v = *(const v4f*)&srows[k * NFEAT + lane * 4];
        __builtin_nontemporal_store(
            v, (v4f*)&outBA[(size_t)p * NFEAT + lane * 4]);
    }
}

extern "C" void kernel_launch(void* const* d_in, const int* in_sizes, int n_in,
                              void* d_out, int out_size, void* d_ws, size_t ws_size,
                              hipStream_t stream) {
    const float* emb = (const float*)d_in[0];   // node_embedding (B, AT, 128) f32
    const int*   nbr = (const int*)d_in[1];     // nbr_idx (B, AT, 32) int
    float*       out = (float*)d_out;           // (B, AT, 32, 31, 128) f32

    const int blocks = BATCH * AT;              // 2048 blocks, one per (b, a)
    GetNodeK_gather_kernel<<<blocks, 256, 0, stream>>>(emb, nbr, out);
}